// MambaBlock_26225070309545
// MI455X (gfx1250) — compile-verified
//
#include <hip/hip_runtime.h>
#include <hip/hip_bf16.h>
#include <math.h>
#include <stdint.h>

typedef float v2f __attribute__((ext_vector_type(2)));
typedef float v8f __attribute__((ext_vector_type(8)));

#define B_    1
#define L_    2048
#define DM    768
#define DI    1536
#define DTR   48
#define DS    16
#define DCONV 4
#define LCHUNK 128
#define NCHUNK (L_ / LCHUNK)          // 16
#define NSTATE_THREADS (DI * DS)      // 24576

// LDS-staged GEMM tiling
#define TM 128
#define TN 64
#define TK 16
#define A_STRIDE 20   // padded A tile: bank-conflict-free, rows 16B aligned

__device__ __forceinline__ float silu_f(float x) {
    return x * (1.0f / (1.0f + expf(-x)));
}
__device__ __forceinline__ float softplus_f(float x) {
    return fmaxf(x, 0.0f) + log1pf(expf(-fabsf(x)));
}
__device__ __forceinline__ v8f wmma4(v2f a, v2f b, v8f c) {
    return __builtin_amdgcn_wmma_f32_16x16x4_f32(false, a, false, b, (short)0, c,
                                                 false, false);
}
__device__ __forceinline__ unsigned lds_off(const float* p) {
    return (unsigned)(uintptr_t)p;   // low 32 bits of generic ptr = LDS byte addr
}
__device__ __forceinline__ void async_b128_to_lds(unsigned lds, const float* gaddr) {
    asm volatile("global_load_async_to_lds_b128 %0, %1, off"
                 :: "v"(lds), "v"(gaddr) : "memory");
}

// ---------------------------------------------------------------------------
// LDS-staged fp32 WMMA GEMM with DOUBLE-BUFFERED async global->LDS pipeline.
// Block = 256 threads = 8 waves -> 128x64 tile; wave = 32x32 (2x2 wmma tiles).
// Per chunk: 3 GLOBAL_LOAD_ASYNC_TO_LDS_B128 per thread; next chunk is issued
// into the alternate buffer, then s_wait_asynccnt 3 retires only the current
// chunk's (older, in-order) transfers so the new DMA overlaps WMMA compute.
// V_WMMA_F32_16X16X4_F32 layouts (ISA 7.12.2):
//   A 16x4 : lane m=lane&15, kh=lane>>4 -> {A[m][2kh], A[m][2kh+1]}
//   B 4x16 : lane n=lane&15, kh=lane>>4 -> {B[2kh][n], B[2kh+1][n]}
//   C/D    : vgpr j: lanes 0-15 -> M=j, lanes 16-31 -> M=j+8; N=lane&15
// EPI: 0 = plain store; 1 = split store (col<splitN -> C0 else silu -> C1).
// ---------------------------------------------------------------------------
template <int EPI>
__global__ __launch_bounds__(256) void wmma_gemm_f32_lds(
    const float* __restrict__ A, int lda,
    const float* __restrict__ B, int ldb,
    float* __restrict__ C0, float* __restrict__ C1, int ldc,
    int K, int splitN, int NtilesX)
{
    __shared__ float As[2 * TM * A_STRIDE];
    __shared__ float Bs[2 * TK * TN];

    const int bM   = (blockIdx.x / NtilesX) * TM;
    const int bN   = (blockIdx.x % NtilesX) * TN;
    const int tid  = threadIdx.x;
    const int wave = tid >> 5;
    const int lane = tid & 31;
    const int wm   = wave >> 1;   // 0..3 : 32-row band
    const int wn   = wave & 1;    // 0..1 : 32-col band
    const int m    = lane & 15;
    const int kh   = lane >> 4;

    // async-fill assignment: A: 2 threads/row, 2 x b128 each; B: 1 x b128/thread
    const int ar  = tid >> 1;
    const int asg = (tid & 1) * 8;          // float offset within A row (0 or 8)
    const int br  = tid >> 4;
    const int bsg = (tid & 15) * 4;         // float offset within B row

    const unsigned ldsA0   = lds_off(&As[ar * A_STRIDE + asg]);
    const unsigned ldsB0   = lds_off(&Bs[br * TN + bsg]);
    const unsigned aBufB   = TM * A_STRIDE * 4;
    const unsigned bBufB   = TK * TN * 4;
    const float* gA0 = A + (bM + ar) * lda + asg;
    const float* gB0 = B + br * ldb + bN + bsg;

    v8f acc[2][2] = {};

    // prologue: fill buffer 0 with chunk 0
    async_b128_to_lds(ldsA0,      gA0);
    async_b128_to_lds(ldsA0 + 16, gA0 + 4);
    async_b128_to_lds(ldsB0,      gB0);

    int buf = 0;
    for (int k0 = 0; k0 < K; k0 += TK) {
        if (k0 + TK < K) {
            // issue next chunk into the other buffer (safe: previous iteration's
            // closing barrier means no wave still reads it), then retire only
            // the 3 older transfers (in-order completion per wave).
            const unsigned nb = buf ^ 1;
            async_b128_to_lds(ldsA0      + nb * aBufB, gA0 + (k0 + TK));
            async_b128_to_lds(ldsA0 + 16 + nb * aBufB, gA0 + (k0 + TK) + 4);
            async_b128_to_lds(ldsB0      + nb * bBufB, gB0 + (size_t)(k0 + TK) * ldb);
            asm volatile("s_wait_asynccnt 3" ::: "memory");
        } else {
            asm volatile("s_wait_asynccnt 0" ::: "memory");
        }
        __syncthreads();

        const float* Ab = &As[buf * TM * A_STRIDE];
        const float* Bb = &Bs[buf * TK * TN];
        #pragma unroll
        for (int kk = 0; kk < TK; kk += 4) {
            const int kc = kk + 2 * kh;
            v2f a0 = *(const v2f*)&Ab[(wm * 32 +      m) * A_STRIDE + kc];
            v2f a1 = *(const v2f*)&Ab[(wm * 32 + 16 + m) * A_STRIDE + kc];
            v2f b0, b1;
            b0.x = Bb[ kc      * TN + wn * 32 +      m];
            b0.y = Bb[(kc + 1) * TN + wn * 32 +      m];
            b1.x = Bb[ kc      * TN + wn * 32 + 16 + m];
            b1.y = Bb[(kc + 1) * TN + wn * 32 + 16 + m];
            acc[0][0] = wmma4(a0, b0, acc[0][0]);
            acc[0][1] = wmma4(a0, b1, acc[0][1]);
            acc[1][0] = wmma4(a1, b0, acc[1][0]);
            acc[1][1] = wmma4(a1, b1, acc[1][1]);
        }
        __syncthreads();
        buf ^= 1;
    }

    #pragma unroll
    for (int i = 0; i < 2; ++i) {
        #pragma unroll
        for (int j = 0; j < 2; ++j) {
            const int col = bN + wn * 32 + j * 16 + m;
            #pragma unroll
            for (int v = 0; v < 8; ++v) {
                const int row = bM + wm * 32 + i * 16 + kh * 8 + v;
                float val = acc[i][j][v];
                if (EPI == 1) {
                    if (col < splitN) C0[row * ldc + col] = val;
                    else              C1[row * ldc + (col - splitN)] = silu_f(val);
                } else {
                    C0[row * ldc + col] = val;
                }
            }
        }
    }
}

// ---------------------------------------------------------------------------
// Simple per-wave WMMA GEMM for the small GEMMs (x_proj N=80, dt_proj K=48).
// EPI: 0 = plain store, 2 = +bias then softplus.
// ---------------------------------------------------------------------------
template <int EPI>
__global__ __launch_bounds__(128) void wmma_gemm_f32(
    const float* __restrict__ A, int lda,
    const float* __restrict__ B, int ldb,
    float* __restrict__ C0, int ldc,
    const float* __restrict__ bias,
    int Mtiles, int Ntiles, int K)
{
    const int wave = threadIdx.x >> 5;
    const int lane = threadIdx.x & 31;
    const int tile = blockIdx.x * 4 + wave;
    if (tile >= Mtiles * Ntiles) return;          // wave-uniform
    const int tm = tile / Ntiles;
    const int tn = tile % Ntiles;
    const int m  = lane & 15;
    const int kh = lane >> 4;

    const float* Arow = A + (tm * 16 + m) * lda + 2 * kh;
    const float* Bcol = B + (2 * kh) * ldb + tn * 16 + m;

    v8f acc = {};
    #pragma unroll 4
    for (int k = 0; k < K; k += 4) {
        v2f a, b;
        a.x = Arow[k];
        a.y = Arow[k + 1];
        b.x = Bcol[k * ldb];
        b.y = Bcol[k * ldb + ldb];
        acc = wmma4(a, b, acc);
    }

    const int col = tn * 16 + m;
    #pragma unroll
    for (int j = 0; j < 8; ++j) {
        const int row = tm * 16 + kh * 8 + j;
        float v = acc[j];
        if (EPI == 2) v = softplus_f(v + bias[col]);
        C0[row * ldc + col] = v;
    }
}

// ---------------------------------------------------------------------------
// Depthwise causal conv (width 4) + bias + SiLU.
// ---------------------------------------------------------------------------
__global__ __launch_bounds__(256) void conv_silu_kernel(
    const float* __restrict__ xs_raw,
    const float* __restrict__ conv_w,   // [DCONV,1,DI] flat: t*DI + d
    const float* __restrict__ conv_b,
    float* __restrict__ xs_conv)
{
    const int idx = blockIdx.x * blockDim.x + threadIdx.x;
    if (idx >= L_ * DI) return;
    const int d = idx % DI;
    const int l = idx / DI;
    float acc = conv_b[d];
    #pragma unroll
    for (int t = 0; t < DCONV; ++t) {
        const int ll = l - (DCONV - 1) + t;
        if (ll >= 0) acc += xs_raw[ll * DI + d] * conv_w[t * DI + d];
    }
    xs_conv[idx] = silu_f(acc);
}

// ---------------------------------------------------------------------------
// Chunked selective scan: h_l = alpha_l h_{l-1} + beta_l, alpha=exp(delta*a),
// beta=delta*b*u.  Pass 1: per-chunk P=prod(alpha), S=zero-init result.
// ---------------------------------------------------------------------------
__global__ __launch_bounds__(256) void scan_pass1(
    const float* __restrict__ u, const float* __restrict__ delta,
    const float* __restrict__ xdbl, const float* __restrict__ a_log,
    float* __restrict__ Pbuf, float* __restrict__ Sbuf)
{
    const int t = blockIdx.x * blockDim.x + threadIdx.x;
    if (t >= NCHUNK * NSTATE_THREADS) return;
    const int n  = t & 15;
    const int dc = t >> 4;
    const int d  = dc % DI;
    const int c  = dc / DI;
    const float a = -expf(a_log[d * DS + n]);
    float P = 1.0f, S = 0.0f;
    const int l0 = c * LCHUNK;
    for (int i = 0; i < LCHUNK; ++i) {
        const int l = l0 + i;
        const float dl    = delta[l * DI + d];
        const float alpha = expf(dl * a);
        const float beta  = dl * xdbl[l * 80 + DTR + n] * u[l * DI + d];
        S = alpha * S + beta;
        P *= alpha;
    }
    Pbuf[t] = P;
    Sbuf[t] = S;
}

__global__ __launch_bounds__(256) void scan_pass2(
    const float* __restrict__ Pbuf, const float* __restrict__ Sbuf,
    float* __restrict__ H0)
{
    const int t = blockIdx.x * blockDim.x + threadIdx.x;
    if (t >= NSTATE_THREADS) return;
    float h = 0.0f;
    #pragma unroll
    for (int c = 0; c < NCHUNK; ++c) {
        H0[c * NSTATE_THREADS + t] = h;
        h = Pbuf[c * NSTATE_THREADS + t] * h + Sbuf[c * NSTATE_THREADS + t];
    }
}

__global__ __launch_bounds__(256) void scan_pass3(
    const float* __restrict__ u, const float* __restrict__ delta,
    const float* __restrict__ xdbl, const float* __restrict__ a_log,
    const float* __restrict__ d_param, const float* __restrict__ silu_res,
    const float* __restrict__ H0, float* __restrict__ y2)
{
    const int t = blockIdx.x * blockDim.x + threadIdx.x;
    if (t >= NCHUNK * NSTATE_THREADS) return;
    const int n  = t & 15;
    const int dc = t >> 4;
    const int d  = dc % DI;
    const int c  = dc / DI;
    const float a  = -expf(a_log[d * DS + n]);
    const float dp = d_param[d];
    float h = H0[c * NSTATE_THREADS + d * DS + n];
    const int l0 = c * LCHUNK;
    for (int i = 0; i < LCHUNK; ++i) {
        const int l = l0 + i;
        const float dl = delta[l * DI + d];
        const float ul = u[l * DI + d];
        h = expf(dl * a) * h + dl * xdbl[l * 80 + DTR + n] * ul;
        float y = h * xdbl[l * 80 + DTR + DS + n];
        y += __shfl_xor(y, 1, 32);
        y += __shfl_xor(y, 2, 32);
        y += __shfl_xor(y, 4, 32);
        y += __shfl_xor(y, 8, 32);
        if (n == 0) y2[l * DI + d] = (y + ul * dp) * silu_res[l * DI + d];
    }
}

// ---------------------------------------------------------------------------
extern "C" void kernel_launch(void* const* d_in, const int* in_sizes, int n_in,
                              void* d_out, int out_size, void* d_ws, size_t ws_size,
                              hipStream_t stream)
{
    const float* x          = (const float*)d_in[0];
    const float* in_proj_w  = (const float*)d_in[1];
    const float* conv_w     = (const float*)d_in[2];
    const float* conv_b     = (const float*)d_in[3];
    const float* x_proj_w   = (const float*)d_in[4];
    const float* dt_proj_w  = (const float*)d_in[5];
    const float* dt_proj_b  = (const float*)d_in[6];
    const float* a_log      = (const float*)d_in[7];
    const float* d_param    = (const float*)d_in[8];
    const float* out_proj_w = (const float*)d_in[9];
    float* out = (float*)d_out;

    float* ws       = (float*)d_ws;
    float* xs_raw   = ws;                      // L*DI
    float* silu_res = xs_raw   + L_ * DI;      // L*DI
    float* xs_conv  = silu_res + L_ * DI;      // L*DI
    float* x_dbl    = xs_conv  + L_ * DI;      // L*80
    float* delta_b  = x_dbl    + L_ * 80;      // L*DI
    float* Pbuf     = delta_b  + L_ * DI;      // NCHUNK*DI*DS
    float* Sbuf     = Pbuf     + NCHUNK * NSTATE_THREADS;
    float* H0       = Sbuf     + NCHUNK * NSTATE_THREADS;
    float* y2       = xs_raw;                  // reuse (xs_raw dead after conv)

    // 1) in_proj: [L,DM]@[DM,2*DI] -> xs_raw | silu(res)  (async double-buffer)
    {
        const int NtX = (2 * DI) / TN;                       // 48
        wmma_gemm_f32_lds<1><<<(L_ / TM) * NtX, 256, 0, stream>>>(
            x, DM, in_proj_w, 2 * DI, xs_raw, silu_res, DI, DM, DI, NtX);
    }
    // 2) depthwise causal conv + bias + silu
    conv_silu_kernel<<<(L_ * DI + 255) / 256, 256, 0, stream>>>(
        xs_raw, conv_w, conv_b, xs_conv);
    // 3) x_dbl = xs_conv @ x_proj_w : [L,DI]@[DI,80]
    {
        const int Mt = L_ / 16, Nt = 80 / 16;
        wmma_gemm_f32<0><<<(Mt * Nt + 3) / 4, 128, 0, stream>>>(
            xs_conv, DI, x_proj_w, 80, x_dbl, 80, nullptr, Mt, Nt, DI);
    }
    // 4) delta = softplus(x_dbl[:, :DTR] @ dt_proj_w + dt_proj_b)
    {
        const int Mt = L_ / 16, Nt = DI / 16;
        wmma_gemm_f32<2><<<(Mt * Nt + 3) / 4, 128, 0, stream>>>(
            x_dbl, 80, dt_proj_w, DI, delta_b, DI, dt_proj_b, Mt, Nt, DTR);
    }
    // 5) chunked selective scan
    scan_pass1<<<(NCHUNK * NSTATE_THREADS + 255) / 256, 256, 0, stream>>>(
        xs_conv, delta_b, x_dbl, a_log, Pbuf, Sbuf);
    scan_pass2<<<(NSTATE_THREADS + 255) / 256, 256, 0, stream>>>(Pbuf, Sbuf, H0);
    scan_pass3<<<(NCHUNK * NSTATE_THREADS + 255) / 256, 256, 0, stream>>>(
        xs_conv, delta_b, x_dbl, a_log, d_param, silu_res, H0, y2);
    // 6) out = y2 @ out_proj_w : [L,DI]@[DI,DM]  (async double-buffer)
    {
        const int NtX = DM / TN;                             // 12
        wmma_gemm_f32_lds<0><<<(L_ / TM) * NtX, 256, 0, stream>>>(
            y2, DI, out_proj_w, DM, out, nullptr, DM, DI, 0, NtX);
    }
}